// TGCN_28303834480676
// MI455X (gfx1250) — compile-verified
//
#include <hip/hip_runtime.h>
#include <hip/hip_bf16.h>

// ---------------- problem constants (from reference) ----------------
#define NN        50000
#define EE        800000
#define ETOT      (EE + NN)      // edges + self loops
#define FIN       64
#define HDIM      128
#define GG        512
#define NEGSLOPE  0.2f

typedef __attribute__((ext_vector_type(2))) float v2f;
typedef __attribute__((ext_vector_type(8))) float v8f;

// ---- order-preserving float<->uint encoding for atomic max on floats ----
__device__ __forceinline__ unsigned ord_enc(float f) {
  unsigned u = __float_as_uint(f);
  return (u & 0x80000000u) ? ~u : (u | 0x80000000u);
}
__device__ __forceinline__ float ord_dec(unsigned u) {
  return __uint_as_float((u & 0x80000000u) ? (u & 0x7FFFFFFFu) : ~u);
}

__device__ __forceinline__ float lrelu(float v) {
  return fmaxf(v, 0.0f) + NEGSLOPE * fminf(v, 0.0f);
}

// ---- CDNA5 async global->LDS copy (ASYNCcnt path), 16B per lane ----
__device__ __forceinline__ void async_copy_b128(void* lds_ptr, const void* gptr) {
  unsigned lds_off = (unsigned)(size_t)lds_ptr;   // low 32 bits of LDS aperture = LDS byte addr
  asm volatile("global_load_async_to_lds_b128 %0, %1, off"
               :: "v"(lds_off), "v"(gptr)
               : "memory");
}
__device__ __forceinline__ void wait_async0() {
  asm volatile("s_wait_asynccnt 0" ::: "memory");
}

// ---------------- zero fill ----------------
__global__ void fill_zero_k(float* __restrict__ p, long n) {
  long i = (long)blockIdx.x * blockDim.x + threadIdx.x;
  if (i < n) p[i] = 0.0f;
}

// ---------------- fp32 WMMA GEMM with async-LDS staging ----------------
// Y[nrows,HDIM] = X[nrows,K] @ W[K,HDIM] + bias.
// One block (8 waves, 256 thr) computes a 16-row x 128-col output strip.
// K is chunked by KC=32 through LDS (async b128 loads), each wave owns one
// 16x16 column tile and runs 8 x V_WMMA_F32_16X16X4_F32 per chunk.
// A (16x4 f32) frag: lanes 0-15 -> M=lane, K0/K1; lanes 16-31 -> M=lane-16, K2/K3.
// C/D (16x16 f32): VGPR j -> M = j + (lane<16 ? 0 : 8), N = lane&15.
#define KC 32
__global__ __launch_bounds__(256) void gemm_bias_k(const float* __restrict__ X,
                                                   const float* __restrict__ W,
                                                   const float* __restrict__ bias,
                                                   float* __restrict__ Y,
                                                   int nrows, int K) {
  __shared__ float As[16 * KC];        // 2 KB: 16 rows x 32 k
  __shared__ float Bs[KC * HDIM];      // 16 KB: 32 k x 128 cols

  const int tid  = threadIdx.x;
  const int wave = tid >> 5;           // 0..7 -> column tile tn
  const int lane = tid & 31;
  const int half = lane >> 4;
  const int l15  = lane & 15;
  const int m0   = blockIdx.x * 16;    // nrows % 16 == 0, no remainder

  v8f acc = {};

  for (int k0 = 0; k0 < K; k0 += KC) {
    // ---- stage A chunk: 16 rows x 32 floats = 128 float4 ----
    for (int q = tid; q < 16 * KC / 4; q += 256) {
      int r  = q >> 3;                 // row 0..15
      int kq = q & 7;                  // float4 within row
      async_copy_b128(&As[r * KC + kq * 4],
                      X + (size_t)(m0 + r) * K + k0 + kq * 4);
    }
    // ---- stage B chunk: 32 k-rows x 128 floats = 1024 float4 ----
    for (int q = tid; q < KC * HDIM / 4; q += 256) {
      int kk = q >> 5;                 // k-row 0..31
      int cq = q & 31;                 // float4 within row
      async_copy_b128(&Bs[kk * HDIM + cq * 4],
                      W + (size_t)(k0 + kk) * HDIM + cq * 4);
    }
    wait_async0();
    __syncthreads();

    // ---- 8 WMMA k-steps over this chunk ----
#pragma unroll
    for (int ks = 0; ks < KC; ks += 4) {
      int ka0 = ks + (half ? 2 : 0);
      int ka1 = ks + (half ? 3 : 1);
      v2f a, b;
      a.x = As[l15 * KC + ka0];
      a.y = As[l15 * KC + ka1];
      b.x = Bs[ka0 * HDIM + wave * 16 + l15];
      b.y = Bs[ka1 * HDIM + wave * 16 + l15];
      acc = __builtin_amdgcn_wmma_f32_16x16x4_f32(false, a, false, b,
                                                  (short)0, acc, false, false);
    }
    __syncthreads();
  }

  const int col   = wave * 16 + l15;
  const float bv  = bias[col];
  const int mbase = m0 + (half ? 8 : 0);
#pragma unroll
  for (int j = 0; j < 8; ++j) {
    Y[(size_t)(mbase + j) * HDIM + col] = acc[j] + bv;
  }
}

// ---------------- edge logits + segment max ----------------
// One wave per edge; lane l holds features [4l, 4l+4).
__global__ void edge_logits_k(const float* __restrict__ xl,
                              const float* __restrict__ xr,
                              const float* __restrict__ att,
                              const int* __restrict__ ei,
                              float* __restrict__ logits,
                              unsigned* __restrict__ m_enc) {
  int e = blockIdx.x * (blockDim.x >> 5) + (threadIdx.x >> 5);
  if (e >= ETOT) return;
  int lane = threadIdx.x & 31;
  int src, dst;
  if (e < EE) { src = ei[e]; dst = ei[EE + e]; }
  else        { src = e - EE; dst = e - EE; }

  const float4* pl = (const float4*)(xl + (size_t)src * HDIM);
  const float4* pr = (const float4*)(xr + (size_t)dst * HDIM);
  const float4* pa = (const float4*)att;
  float4 a = pl[lane];
  float4 b = pr[lane];
  float4 w = pa[lane];
  float s = w.x * lrelu(a.x + b.x) + w.y * lrelu(a.y + b.y) +
            w.z * lrelu(a.z + b.z) + w.w * lrelu(a.w + b.w);
#pragma unroll
  for (int off = 16; off > 0; off >>= 1) s += __shfl_xor(s, off, 32);

  if (lane == 0) {
    logits[e] = s;
    atomicMax(&m_enc[dst], ord_enc(s));
  }
}

// ---------------- exp(logit - max) + segment sum ----------------
__global__ void edge_exp_k(float* __restrict__ logits,
                           const unsigned* __restrict__ m_enc,
                           const int* __restrict__ ei,
                           float* __restrict__ denom) {
  int e = blockIdx.x * blockDim.x + threadIdx.x;
  if (e >= ETOT) return;
  int dst = (e < EE) ? ei[EE + e] : (e - EE);
  float m  = ord_dec(m_enc[dst]);
  float ea = __expf(logits[e] - m);
  logits[e] = ea;
  atomicAdd(&denom[dst], ea);
}

// ---------------- weighted aggregation: agg[dst] += alpha * xl[src] ----------------
__global__ void edge_agg_k(const float* __restrict__ xl,
                           const float* __restrict__ logits,
                           const float* __restrict__ denom,
                           const int* __restrict__ ei,
                           float* __restrict__ agg) {
  int e = blockIdx.x * (blockDim.x >> 5) + (threadIdx.x >> 5);
  if (e >= ETOT) return;
  int lane = threadIdx.x & 31;
  int src, dst;
  if (e < EE) { src = ei[e]; dst = ei[EE + e]; }
  else        { src = e - EE; dst = e - EE; }

  float alpha = logits[e] / denom[dst];
  float4 v = ((const float4*)(xl + (size_t)src * HDIM))[lane];
  float* base = agg + (size_t)dst * HDIM + lane * 4;
  atomicAdd(base + 0, v.x * alpha);
  atomicAdd(base + 1, v.y * alpha);
  atomicAdd(base + 2, v.z * alpha);
  atomicAdd(base + 3, v.w * alpha);
}

// ---------------- h = relu(agg + bias) ----------------
__global__ void bias_relu_k(const float* __restrict__ agg,
                            const float* __restrict__ bias,
                            float* __restrict__ h, long n) {
  long i = (long)blockIdx.x * blockDim.x + threadIdx.x;
  if (i < n) h[i] = fmaxf(agg[i] + bias[i & (HDIM - 1)], 0.0f);
}

// ---------------- pooling ----------------
__global__ void pool_feat_k(const float* __restrict__ h,
                            const int* __restrict__ batch,
                            float* __restrict__ sums) {
  long i = (long)blockIdx.x * blockDim.x + threadIdx.x;
  if (i >= (long)NN * HDIM) return;
  int n = (int)(i >> 7);
  int f = (int)(i & (HDIM - 1));
  atomicAdd(&sums[(size_t)batch[n] * HDIM + f], h[i]);
}

__global__ void pool_cnt_k(const int* __restrict__ batch,
                           float* __restrict__ cnt) {
  int n = blockIdx.x * blockDim.x + threadIdx.x;
  if (n < NN) atomicAdd(&cnt[batch[n]], 1.0f);
}

// ---------------- head: out[g] = (sums[g]/cnt[g]) . Wlin + blin ----------------
__global__ void head_k(const float* __restrict__ sums,
                       const float* __restrict__ cnt,
                       const float* __restrict__ Wlin,
                       const float* __restrict__ blin,
                       float* __restrict__ out) {
  int g = blockIdx.x * (blockDim.x >> 5) + (threadIdx.x >> 5);
  if (g >= GG) return;
  int lane = threadIdx.x & 31;
  float4 s = ((const float4*)(sums + (size_t)g * HDIM))[lane];
  float4 w = ((const float4*)Wlin)[lane];
  float p = s.x * w.x + s.y * w.y + s.z * w.z + s.w * w.w;
#pragma unroll
  for (int off = 16; off > 0; off >>= 1) p += __shfl_xor(p, off, 32);
  if (lane == 0) out[g] = p / fmaxf(cnt[g], 1.0f) + blin[0];
}

// =======================================================================
extern "C" void kernel_launch(void* const* d_in, const int* in_sizes, int n_in,
                              void* d_out, int out_size, void* d_ws, size_t ws_size,
                              hipStream_t stream) {
  const float* x     = (const float*)d_in[0];
  const int*   ei    = (const int*)d_in[1];
  // d_in[2] edge_weight: unused by GATv2
  const int*   batch = (const int*)d_in[3];
  const float* Wl1 = (const float*)d_in[4];
  const float* bl1 = (const float*)d_in[5];
  const float* Wr1 = (const float*)d_in[6];
  const float* br1 = (const float*)d_in[7];
  const float* att1 = (const float*)d_in[8];
  const float* b1   = (const float*)d_in[9];
  const float* Wl2 = (const float*)d_in[10];
  const float* bl2 = (const float*)d_in[11];
  const float* Wr2 = (const float*)d_in[12];
  const float* br2 = (const float*)d_in[13];
  const float* att2 = (const float*)d_in[14];
  const float* b2   = (const float*)d_in[15];
  const float* Wlin = (const float*)d_in[16];
  const float* blin = (const float*)d_in[17];
  float* out = (float*)d_out;

  // ---- workspace layout (floats) ----
  float* ws     = (float*)d_ws;
  const long NH = (long)NN * HDIM;
  float*    xl     = ws;
  float*    xr     = xl + NH;
  float*    agg    = xr + NH;
  float*    h      = agg + NH;
  float*    logits = h + NH;
  unsigned* m_enc  = (unsigned*)(logits + ETOT);
  float*    denom  = (float*)(m_enc + NN);
  float*    sums   = denom + NN;
  float*    cnt    = sums + (long)GG * HDIM;

  auto zero = [&](float* p, long n) {
    fill_zero_k<<<dim3((unsigned)((n + 255) / 256)), dim3(256), 0, stream>>>(p, n);
  };

  const dim3 gemm_grid(NN / 16), gemm_blk(256);            // 3125 blocks, 8 waves each
  const dim3 ewave_grid((ETOT + 7) / 8), ewave_blk(256);   // 8 edges/block (wave each)
  const dim3 ethr_grid((ETOT + 255) / 256), ethr_blk(256);
  const dim3 nh_grid((unsigned)((NH + 255) / 256)), nh_blk(256);

  // =================== layer 1 ===================
  gemm_bias_k<<<gemm_grid, gemm_blk, 0, stream>>>(x, Wl1, bl1, xl, NN, FIN);
  gemm_bias_k<<<gemm_grid, gemm_blk, 0, stream>>>(x, Wr1, br1, xr, NN, FIN);
  zero((float*)m_enc, NN);          // 0u < ord_enc(-inf), a lower bound
  zero(denom, NN);
  zero(agg, NH);
  edge_logits_k<<<ewave_grid, ewave_blk, 0, stream>>>(xl, xr, att1, ei, logits, m_enc);
  edge_exp_k<<<ethr_grid, ethr_blk, 0, stream>>>(logits, m_enc, ei, denom);
  edge_agg_k<<<ewave_grid, ewave_blk, 0, stream>>>(xl, logits, denom, ei, agg);
  bias_relu_k<<<nh_grid, nh_blk, 0, stream>>>(agg, b1, h, NH);

  // =================== layer 2 ===================
  gemm_bias_k<<<gemm_grid, gemm_blk, 0, stream>>>(h, Wl2, bl2, xl, NN, HDIM);
  gemm_bias_k<<<gemm_grid, gemm_blk, 0, stream>>>(h, Wr2, br2, xr, NN, HDIM);
  zero((float*)m_enc, NN);
  zero(denom, NN);
  zero(agg, NH);
  edge_logits_k<<<ewave_grid, ewave_blk, 0, stream>>>(xl, xr, att2, ei, logits, m_enc);
  edge_exp_k<<<ethr_grid, ethr_blk, 0, stream>>>(logits, m_enc, ei, denom);
  edge_agg_k<<<ewave_grid, ewave_blk, 0, stream>>>(xl, logits, denom, ei, agg);
  bias_relu_k<<<nh_grid, nh_blk, 0, stream>>>(agg, b2, h, NH);

  // =================== pool + head ===================
  zero(sums, (long)GG * HDIM);
  zero(cnt, GG);
  pool_feat_k<<<nh_grid, nh_blk, 0, stream>>>(h, batch, sums);
  pool_cnt_k<<<dim3((NN + 255) / 256), dim3(256), 0, stream>>>(batch, cnt);
  head_k<<<dim3((GG + 7) / 8), dim3(256), 0, stream>>>(sums, cnt, Wlin, blin, out);
}